// TransductiveLayer_43250320671016
// MI455X (gfx1250) — compile-verified
//
#include <hip/hip_runtime.h>

#define N_NODES 100000
#define DIM     256
#define NHOPS   3
#define NEDGES  3200000

typedef float v2f __attribute__((ext_vector_type(2)));
typedef float v4f __attribute__((ext_vector_type(4)));
typedef float v8f __attribute__((ext_vector_type(8)));

// ---------------- zero / relu passes (float4 vectorized) ----------------

__global__ void __launch_bounds__(256) zero_kernel(v4f* __restrict__ out) {
  size_t i = (size_t)blockIdx.x * blockDim.x + threadIdx.x;
  v4f z = {0.f, 0.f, 0.f, 0.f};
  out[i] = z;
}

__global__ void __launch_bounds__(256) relu_kernel(v4f* __restrict__ out) {
  size_t i = (size_t)blockIdx.x * blockDim.x + threadIdx.x;
  v4f x = out[i];  // final pass: out is still L2-hot from the atomics
  x.x = fmaxf(x.x, 0.f);
  x.y = fmaxf(x.y, 0.f);
  x.z = fmaxf(x.z, 0.f);
  x.w = fmaxf(x.w, 0.f);
  __builtin_nontemporal_store(x, &out[i]);  // last touch: don't keep in cache
}

// ---------------- dense GEMM: F = X @ W, fp32 WMMA 16x16x4 ----------------
// Wave computes a 16x64 output strip. Layouts per CDNA5 ISA 7.12.2:
//   A 16x4 fp32 : lane(0-15)=M, VGPR0/1 = K={0,1} (lanes 0-15) / K={2,3} (lanes 16-31)
//   B 4x16 fp32 : lanes hold N; VGPR0/1 = K rows {2*hi, 2*hi+1}
//   C/D 16x16   : VGPR v -> row M = v + 8*hi, col N = lane&15

__global__ void __launch_bounds__(256) gemm_kernel(const float* __restrict__ X,
                                                   const float* __restrict__ W,
                                                   float* __restrict__ F) {
  const int wave = threadIdx.x >> 5;
  const int lane = threadIdx.x & 31;
  const int lo   = lane & 15;
  const int hi   = lane >> 4;

  const int row0 = blockIdx.x * 32 + (wave >> 2) * 16;  // 2 row-tiles per block
  const int col0 = (wave & 3) * 64;                      // 4 col-groups per block

  v8f acc0 = {}, acc1 = {}, acc2 = {}, acc3 = {};

  const float* xrow = X + (size_t)(row0 + lo) * DIM;

  for (int k0 = 0; k0 < DIM; k0 += 4) {
    const int ka = k0 + 2 * hi;                 // this half-wave's K rows
    v2f a = *(const v2f*)(xrow + ka);           // A frag: X[row][ka], X[row][ka+1]

    const float* w0 = W + (size_t)ka * DIM + col0 + lo;  // K row ka
    const float* w1 = w0 + DIM;                          // K row ka+1

    v2f b0 = { w0[0],  w1[0]  };
    v2f b1 = { w0[16], w1[16] };
    v2f b2 = { w0[32], w1[32] };
    v2f b3 = { w0[48], w1[48] };

    acc0 = __builtin_amdgcn_wmma_f32_16x16x4_f32(false, a, false, b0, (short)0, acc0, false, false);
    acc1 = __builtin_amdgcn_wmma_f32_16x16x4_f32(false, a, false, b1, (short)0, acc1, false, false);
    acc2 = __builtin_amdgcn_wmma_f32_16x16x4_f32(false, a, false, b2, (short)0, acc2, false, false);
    acc3 = __builtin_amdgcn_wmma_f32_16x16x4_f32(false, a, false, b3, (short)0, acc3, false, false);
  }

  float* frow = F + (size_t)(row0 + 8 * hi) * DIM + col0 + lo;
#pragma unroll
  for (int v = 0; v < 8; ++v) {
    float* fr = frow + (size_t)v * DIM;
    fr[0]  = acc0[v];
    fr[16] = acc1[v];
    fr[32] = acc2[v];
    fr[48] = acc3[v];
  }
}

// ---------------- COO SpMM scatter: out[r] += val * F[c] ----------------
// Each wave batch-loads 32 edges with coalesced NT loads (edge lists are
// streamed once -> TH=NT keeps feats/out resident in the 192MB L2), then
// broadcasts each edge via __shfl (wave32 ds_bpermute). Per edge: 2x b128
// gathers of the 1KB feature row + 8x global_atomic_add_f32 scatters/lane.

__global__ void __launch_bounds__(256) spmm_kernel(const float* __restrict__ vals,
                                                   const int*   __restrict__ rows,
                                                   const int*   __restrict__ cols,
                                                   const float* __restrict__ F,
                                                   float*       __restrict__ out) {
  const int wave  = threadIdx.x >> 5;
  const int lane  = threadIdx.x & 31;
  const int ebase = (blockIdx.x * 8 + wave) * 32;

  const float ve = __builtin_nontemporal_load(vals + ebase + lane);
  const int   re = __builtin_nontemporal_load(rows + ebase + lane);
  const int   ce = __builtin_nontemporal_load(cols + ebase + lane);

  for (int j = 0; j < 32; ++j) {
    const float v = __shfl(ve, j, 32);
    const int   r = __shfl(re, j, 32);
    const int   c = __shfl(ce, j, 32);

    const v4f* src = (const v4f*)(F + (size_t)c * DIM + lane * 8);
    v4f f0 = src[0];
    v4f f1 = src[1];

    float* o = out + (size_t)r * DIM + lane * 8;
    unsafeAtomicAdd(o + 0, v * f0.x);
    unsafeAtomicAdd(o + 1, v * f0.y);
    unsafeAtomicAdd(o + 2, v * f0.z);
    unsafeAtomicAdd(o + 3, v * f0.w);
    unsafeAtomicAdd(o + 4, v * f1.x);
    unsafeAtomicAdd(o + 5, v * f1.y);
    unsafeAtomicAdd(o + 6, v * f1.z);
    unsafeAtomicAdd(o + 7, v * f1.w);
  }
}

// ---------------- launch ----------------

extern "C" void kernel_launch(void* const* d_in, const int* in_sizes, int n_in,
                              void* d_out, int out_size, void* d_ws, size_t ws_size,
                              hipStream_t stream) {
  const float* X    = (const float*)d_in[0];  // [N, D]
  const float* Wt   = (const float*)d_in[1];  // [HOPS, D, D]
  const float* vals = (const float*)d_in[2];  // [HOPS, E]
  const int*   rows = (const int*)d_in[3];    // [HOPS, E]
  const int*   cols = (const int*)d_in[4];    // [HOPS, E]
  float* out   = (float*)d_out;               // [N, D]
  float* feats = (float*)d_ws;                // [N, D] scratch, reused per hop

  dim3 blk(256);
  const int nd4_blocks = (N_NODES * DIM) / (256 * 4);  // 25000

  zero_kernel<<<nd4_blocks, blk, 0, stream>>>((v4f*)out);

  for (int hop = 0; hop < NHOPS; ++hop) {
    gemm_kernel<<<N_NODES / 32, blk, 0, stream>>>(
        X, Wt + (size_t)hop * DIM * DIM, feats);
    spmm_kernel<<<NEDGES / 256, blk, 0, stream>>>(   // 12500 blocks
        vals + (size_t)hop * NEDGES,
        rows + (size_t)hop * NEDGES,
        cols + (size_t)hop * NEDGES,
        feats, out);
  }

  relu_kernel<<<nd4_blocks, blk, 0, stream>>>((v4f*)out);
}